// LightningAttention_39805756899525
// MI455X (gfx1250) — compile-verified
//
#include <hip/hip_runtime.h>
#include <math.h>

typedef float v2f __attribute__((ext_vector_type(2)));
typedef float v8f __attribute__((ext_vector_type(8)));

#define WMMA_F32(a, b, c) \
    __builtin_amdgcn_wmma_f32_16x16x4_f32(false, (a), false, (b), (short)0, (c), false, false)

constexpr int H  = 8;
constexpr int D  = 16;
constexpr int L  = 16384;
constexpr int B  = 8;
constexpr int C  = 128;     // channels = H*D
constexpr int CH = 128;     // chunk length
constexpr int NC = L / CH;  // 128 chunks per sequence

__device__ __forceinline__ float head_slope(int h) {
    // _build_slopes for H=8: start = 0.5, slope_i = 0.5^(i+1)
    return exp2f(-(float)(h + 1));
}

// ---------------------------------------------------------------------------
// Kernel 1: qkv = (x + pos_bias) @ W_qkv    M=131072, K=128, N=384 (f32 WMMA)
// ---------------------------------------------------------------------------
__global__ __launch_bounds__(256) void qkv_gemm(const float* __restrict__ x,
                                                const float* __restrict__ pos,
                                                const float* __restrict__ Wqkv,
                                                float* __restrict__ qkv) {
    const int lane = threadIdx.x & 31;
    const int wave = threadIdx.x >> 5;
    const int lm = lane & 15, lh = lane >> 4;
    const int NT = (3 * C) / 16;  // 24 col tiles
    int tile = blockIdx.x * 8 + wave;
    int tm = tile / NT, tn = tile % NT;
    int row0 = tm * 16, col0 = tn * 16;
    int trow = row0 & (L - 1);  // token index (row0 % L)

    const float* xr = x + (size_t)(row0 + lm) * C;
    const float* pr = pos + (size_t)(trow + lm) * C;

    v8f acc = {};
#pragma unroll
    for (int kk = 0; kk < C / 4; ++kk) {
        int k0 = kk * 4 + lh * 2;
        v2f a = *reinterpret_cast<const v2f*>(xr + k0);
        v2f p = *reinterpret_cast<const v2f*>(pr + k0);
        a += p;
        v2f bm;
        bm.x = Wqkv[(size_t)k0 * (3 * C) + col0 + lm];
        bm.y = Wqkv[(size_t)(k0 + 1) * (3 * C) + col0 + lm];
        acc = WMMA_F32(a, bm, acc);
    }
    float* outr = qkv + (size_t)row0 * (3 * C) + col0 + lm;
#pragma unroll
    for (int r = 0; r < 8; ++r)
        outr[(size_t)(r + 8 * lh) * (3 * C)] = acc[r];
}

// ---------------------------------------------------------------------------
// Kernel 2: per-head RMS-norm of q,k (+scale, gamma); repack to [w][b][h][L][D]
// ---------------------------------------------------------------------------
__global__ __launch_bounds__(256) void norm_pack(const float* __restrict__ qkv,
                                                 const float* __restrict__ qg,
                                                 const float* __restrict__ kg,
                                                 float* __restrict__ qkvn) {
    int tid = blockIdx.x * 256 + threadIdx.x;  // (w*B + b)*H*L + h*L + t
    int t = tid & (L - 1);
    int r1 = tid >> 14;
    int h = r1 & (H - 1);
    int r2 = r1 >> 3;
    int b = r2 & (B - 1);
    int w = r2 >> 3;  // 0=q 1=k 2=v

    const float* src = qkv + (size_t)(b * L + t) * (3 * C) + w * C + h * D;
    float vals[16];
#pragma unroll
    for (int e = 0; e < 16; ++e) vals[e] = src[e];

    if (w < 2) {
        const float* g = (w == 0 ? qg : kg) + h * D;
        float scale = (w == 0) ? 0.25f : 1.0f;  // d^-0.5 = 1/4 (cancels in norm)
        float ss = 0.f;
#pragma unroll
        for (int e = 0; e < 16; ++e) { vals[e] *= scale; ss += vals[e] * vals[e]; }
        float n = fmaxf(sqrtf(ss), 1e-12f);
        float inv = 4.0f / n;  // sqrt(d) = 4
#pragma unroll
        for (int e = 0; e < 16; ++e) vals[e] = vals[e] * inv * g[e];
    }
    float* dst = qkvn + ((size_t)((w * B + b) * H + h) * L + t) * D;
#pragma unroll
    for (int e = 0; e < 16; ++e) dst[e] = vals[e];
}

// ---------------------------------------------------------------------------
// Kernel 3a: per-chunk KV_c = (k * k_decay)^T @ v   (16x16, K=128; 1 wave/chunk)
// ---------------------------------------------------------------------------
__global__ __launch_bounds__(256) void chunk_kv(const float* __restrict__ qkvn,
                                                float* __restrict__ kv) {
    const int lane = threadIdx.x & 31;
    const int wave = threadIdx.x >> 5;
    const int lm = lane & 15, lh = lane >> 4;
    int id = blockIdx.x * 8 + wave;       // bh*NC + c0
    int bh = id >> 7, c0 = id & (NC - 1);
    float s = head_slope(bh & (H - 1));

    const float* kbase = qkvn + ((size_t)(1 * B * H + bh) * L + c0 * CH) * D;
    const float* vbase = qkvn + ((size_t)(2 * B * H + bh) * L + c0 * CH) * D;

    v8f acc = {};
#pragma unroll 8
    for (int kk = 0; kk < CH / 4; ++kk) {
        int j0 = kk * 4 + lh * 2;
        float kd0 = expf(-s * (float)(CH - 1 - j0));
        float kd1 = expf(-s * (float)(CH - 2 - j0));
        v2f a;
        a.x = kbase[(size_t)j0 * D + lm] * kd0;       // A[e=lm, j] = k[j][e]*kd(j)
        a.y = kbase[(size_t)(j0 + 1) * D + lm] * kd1;
        v2f bm;
        bm.x = vbase[(size_t)j0 * D + lm];            // B[j, f=lm] = v[j][f]
        bm.y = vbase[(size_t)(j0 + 1) * D + lm];
        acc = WMMA_F32(a, bm, acc);
    }
    float* out = kv + (size_t)id * 256 + lm;
#pragma unroll
    for (int r = 0; r < 8; ++r) out[(r + 8 * lh) * 16] = acc[r];
}

// ---------------------------------------------------------------------------
// Kernel 3b: decayed prefix scan of states: S_c = blk*S_{c-1} + KV_{c-1}
// (stores state ENTERING each chunk); 1 thread per S element, 64 blocks
// ---------------------------------------------------------------------------
__global__ __launch_bounds__(256) void state_scan(const float* __restrict__ kv,
                                                  float* __restrict__ Sst) {
    int bh = blockIdx.x;
    int ef = threadIdx.x;  // e*16 + f
    float blk = expf(-head_slope(bh & (H - 1)) * (float)CH);
    const float* kvb = kv + (size_t)bh * NC * 256 + ef;
    float* so = Sst + (size_t)bh * NC * 256 + ef;
    float S = 0.f;
    for (int c = 0; c < NC; ++c) {
        so[c * 256] = S;
        S = blk * S + kvb[c * 256];
    }
}

// ---------------------------------------------------------------------------
// Kernel 3c: o = mask(q k^T) @ v + (q*q_decay) @ S   (1 block per chunk)
// ---------------------------------------------------------------------------
__global__ __launch_bounds__(256) void chunk_out(const float* __restrict__ qkvn,
                                                 const float* __restrict__ Sst,
                                                 float* __restrict__ o) {
    __shared__ float sc[8][16][20];  // per-wave 16x16 scores tile, padded
    const int lane = threadIdx.x & 31;
    const int wave = threadIdx.x >> 5;
    const int lm = lane & 15, lh = lane >> 4;
    int id = blockIdx.x;  // bh*NC + c0
    int bh = id >> 7, c0 = id & (NC - 1);
    float s = head_slope(bh & (H - 1));

    const float* qbase = qkvn + ((size_t)bh * L + c0 * CH + wave * 16) * D;
    const float* kbase = qkvn + ((size_t)(1 * B * H + bh) * L + c0 * CH) * D;
    const float* vbase = qkvn + ((size_t)(2 * B * H + bh) * L + c0 * CH) * D;
    const float* Sb = Sst + (size_t)id * 256;

    // q fragments: A[m=lm, e=4kk+2lh+{0,1}] (reused for inter + all 8 col tiles)
    v2f qa[4];
#pragma unroll
    for (int kk = 0; kk < 4; ++kk)
        qa[kk] = *reinterpret_cast<const v2f*>(qbase + (size_t)lm * D + kk * 4 + lh * 2);

    // inter-chunk: oacc = (q * q_decay(i)) @ S,  i = row within chunk
    float qdec = expf(-s * (float)(wave * 16 + lm + 1));
    v8f oacc = {};
#pragma unroll
    for (int kk = 0; kk < 4; ++kk) {
        int e0 = kk * 4 + lh * 2;
        v2f a = qa[kk] * qdec;
        v2f bm;
        bm.x = Sb[e0 * 16 + lm];
        bm.y = Sb[(e0 + 1) * 16 + lm];
        oacc = WMMA_F32(a, bm, oacc);
    }

    // intra-chunk: per 16-col tile: scores -> mask -> LDS relayout -> @ v
#pragma unroll
    for (int ct = 0; ct < 8; ++ct) {
        v8f sacc = {};
#pragma unroll
        for (int kk = 0; kk < 4; ++kk) {
            int e0 = kk * 4 + lh * 2;
            // B[e, j=lm] = k[ct*16+lm][e] : contiguous float2 in k row
            v2f bm = *reinterpret_cast<const v2f*>(kbase + (size_t)(ct * 16 + lm) * D + e0);
            sacc = WMMA_F32(qa[kk], bm, sacc);
        }
#pragma unroll
        for (int r = 0; r < 8; ++r) {
            int i = wave * 16 + r + 8 * lh;  // query pos in chunk
            int j = ct * 16 + lm;            // key pos in chunk
            float f = (i >= j) ? expf(-s * (float)(i - j)) : 0.0f;
            sc[wave][r + 8 * lh][lm] = sacc[r] * f;
        }
        // oacc += scores_tile @ v[ct*16 : ct*16+16, :]
#pragma unroll
        for (int kk = 0; kk < 4; ++kk) {
            int jl = kk * 4 + lh * 2;
            v2f a = *reinterpret_cast<const v2f*>(&sc[wave][lm][jl]);
            int j0 = ct * 16 + jl;
            v2f bm;
            bm.x = vbase[(size_t)j0 * D + lm];
            bm.y = vbase[(size_t)(j0 + 1) * D + lm];
            oacc = WMMA_F32(a, bm, oacc);
        }
    }

    float* ob = o + ((size_t)bh * L + c0 * CH + wave * 16) * D + lm;
#pragma unroll
    for (int r = 0; r < 8; ++r) ob[(size_t)(r + 8 * lh) * D] = oacc[r];
}

// ---------------------------------------------------------------------------
// Kernel 4: out = gather(o) @ W_proj + b_proj   M=131072, K=128, N=128
// ---------------------------------------------------------------------------
__global__ __launch_bounds__(256) void out_proj(const float* __restrict__ o,
                                                const float* __restrict__ Wp,
                                                const float* __restrict__ bp,
                                                float* __restrict__ out) {
    const int lane = threadIdx.x & 31;
    const int wave = threadIdx.x >> 5;
    const int lm = lane & 15, lh = lane >> 4;
    int tile = blockIdx.x * 8 + wave;
    int tm = tile >> 3, tn = tile & 7;
    int row0 = tm * 16, col0 = tn * 16;
    int row = row0 + lm;
    int b = row >> 14;       // / L
    int t = row & (L - 1);

    v8f acc = {};
#pragma unroll
    for (int kk = 0; kk < 32; ++kk) {
        int k0 = kk * 4 + lh * 2;
        int hh = k0 >> 4, e = k0 & 15;  // never crosses a head boundary
        v2f a = *reinterpret_cast<const v2f*>(
            o + ((size_t)(b * H + hh) * L + t) * D + e);
        v2f bm;
        bm.x = Wp[(size_t)k0 * C + col0 + lm];
        bm.y = Wp[(size_t)(k0 + 1) * C + col0 + lm];
        acc = WMMA_F32(a, bm, acc);
    }
    float bias = bp[col0 + lm];
    float* outr = out + (size_t)row0 * C + col0 + lm;
#pragma unroll
    for (int r = 0; r < 8; ++r)
        outr[(size_t)(r + 8 * lh) * C] = acc[r] + bias;
}

// ---------------------------------------------------------------------------
extern "C" void kernel_launch(void* const* d_in, const int* in_sizes, int n_in,
                              void* d_out, int out_size, void* d_ws, size_t ws_size,
                              hipStream_t stream) {
    const float* x    = (const float*)d_in[0];
    const float* pos  = (const float*)d_in[1];
    const float* Wqkv = (const float*)d_in[2];
    const float* qg   = (const float*)d_in[3];
    const float* kg   = (const float*)d_in[4];
    const float* Wp   = (const float*)d_in[5];
    const float* bp   = (const float*)d_in[6];
    float* out = (float*)d_out;
    float* ws  = (float*)d_ws;

    // workspace layout (floats):
    // [0, 50331648)            : qkv (dead after norm_pack; region reused below)
    // [50331648, 100663296)    : qkvn = normalized q,k + v in [w][bh][L][D]
    // reuse region 0 after norm_pack:
    //   [0, 2097152)           : per-chunk KV states [bh*NC][16][16]
    //   [2097152, 4194304)     : scanned entering-states S [bh*NC][16][16]
    //   [4194304, 20971520)    : o [bh][L][D]
    float* qkv  = ws;
    float* qkvn = ws + 50331648;
    float* kv   = ws;
    float* Sst  = ws + 2097152;
    float* o    = ws + 4194304;

    qkv_gemm <<<24576, 256, 0, stream>>>(x, pos, Wqkv, qkv);
    norm_pack<<<12288, 256, 0, stream>>>(qkv, qg, kg, qkvn);
    chunk_kv <<<1024,  256, 0, stream>>>(qkvn, kv);
    state_scan<<<64,   256, 0, stream>>>(kv, Sst);
    chunk_out<<<8192,  256, 0, stream>>>(qkvn, Sst, o);
    out_proj <<<8192,  256, 0, stream>>>(o, Wp, bp, out);
}